// EntmaxBisect_55327768707607
// MI455X (gfx1250) — compile-verified
//
#include <hip/hip_runtime.h>
#include <cstdint>

// entmax-1.5 bisection, d = 4096, fp32.
// One 256-thread workgroup per row (grid-strided), row resident in registers,
// async HBM->LDS prefetch of the next row (ASYNCcnt), wave32 reductions.

#define D_DIM   4096
#define TPB     256
#define CHUNKS  4                 // 4 x float4 = 16 floats per thread
#define WAVES   (TPB / 32)        // 8 waves per block (wave32)
#define NITER   50

typedef int v4i_t __attribute__((ext_vector_type(4)));
typedef __attribute__((address_space(1))) v4i_t gv4i;   // global
typedef __attribute__((address_space(3))) v4i_t lv4i;   // LDS

// ---------------- wave32 reductions ----------------
__device__ __forceinline__ float waveSum(float v) {
#pragma unroll
  for (int m = 16; m; m >>= 1) v += __shfl_xor(v, m, 32);
  return v;
}
__device__ __forceinline__ float waveMax(float v) {
#pragma unroll
  for (int m = 16; m; m >>= 1) v = fmaxf(v, __shfl_xor(v, m, 32));
  return v;
}

// Block reduction: one barrier per call; caller alternates parity p so the
// next call's writes never race this call's reads (double-buffered slots).
__device__ __forceinline__ float blockSum(float v, float (*red)[WAVES],
                                          int p, int lane, int wav) {
  v = waveSum(v);
  if (lane == 0) red[p][wav] = v;
  __syncthreads();
  float s = 0.0f;
#pragma unroll
  for (int w = 0; w < WAVES; ++w) s += red[p][w];
  return s;   // bitwise identical in every thread -> uniform control flow
}
__device__ __forceinline__ float blockMax(float v, float (*red)[WAVES],
                                          int p, int lane, int wav) {
  v = waveMax(v);
  if (lane == 0) red[p][wav] = v;
  __syncthreads();
  float s = red[p][0];
#pragma unroll
  for (int w = 1; w < WAVES; ++w) s = fmaxf(s, red[p][w]);
  return s;
}

// ---------------- CDNA5 async HBM -> LDS ----------------
__device__ __forceinline__ void asyncLoadRow(const float* __restrict__ g,
                                             float* l, int tid) {
#pragma unroll
  for (int c = 0; c < CHUNKS; ++c) {
    const float* gp = g + c * (TPB * 4) + tid * 4;   // 16B per lane, coalesced
    float*       lp = l + c * (TPB * 4) + tid * 4;
#if __has_builtin(__builtin_amdgcn_global_load_async_to_lds_b128)
    __builtin_amdgcn_global_load_async_to_lds_b128(
        (gv4i*)gp, (lv4i*)lp, 0, 0);
#else
    unsigned loff =
        (unsigned)(unsigned long long)(__attribute__((address_space(3))) char*)lp;
    asm volatile("global_load_async_to_lds_b128 %0, %1, off"
                 :: "v"(loff), "v"(gp) : "memory");
#endif
  }
}

__device__ __forceinline__ void waitAsync0() {
#if __has_builtin(__builtin_amdgcn_s_wait_asynccnt)
  __builtin_amdgcn_s_wait_asynccnt(0);
#else
  asm volatile("s_wait_asynccnt 0" ::: "memory");
#endif
  asm volatile("" ::: "memory");   // keep LDS reads below the wait
}

// ---------------- partial sum of relu(xs - tau)^2 ----------------
// Wave-uniform skip: if no lane in this wave has any element above tau,
// the wave's contribution is exactly 0.0 — skip all VALU work.
__device__ __forceinline__ float relu2Partial(const float4* xs, float tau,
                                              float lmax) {
  float s = 0.0f;
  if (__ballot(lmax > tau) != 0ull) {
#pragma unroll
    for (int c = 0; c < CHUNKS; ++c) {
      float a = fmaxf(xs[c].x - tau, 0.0f); s = fmaf(a, a, s);
      float b = fmaxf(xs[c].y - tau, 0.0f); s = fmaf(b, b, s);
      float g = fmaxf(xs[c].z - tau, 0.0f); s = fmaf(g, g, s);
      float h = fmaxf(xs[c].w - tau, 0.0f); s = fmaf(h, h, s);
    }
  }
  return s;
}

__global__ void __launch_bounds__(TPB)
entmax15_bisect_kernel(const float* __restrict__ X, float* __restrict__ Out,
                       int nrows) {
  __shared__ float rowbuf[2][D_DIM];   // 32 KB double-buffered row staging
  __shared__ float red[2][WAVES];      // double-buffered reduction slots

  const int tid    = threadIdx.x;
  const int lane   = tid & 31;
  const int wav    = tid >> 5;
  const int stride = gridDim.x;

  int row = blockIdx.x;
  if (row < nrows)
    asyncLoadRow(X + (long long)row * D_DIM, rowbuf[0], tid);

  int rb = 0;   // row-buffer parity
  int p  = 0;   // reduction-slot parity (52 reductions/row: even, stays in sync)

  for (; row < nrows; row += stride, rb ^= 1) {
    waitAsync0();   // prefetch of this row complete

    // LDS -> registers, pre-scaled by (alpha-1)=0.5; track per-thread max.
    float4 xs[CHUNKS];
    float lmax = -3.402823466e+38f;
#pragma unroll
    for (int c = 0; c < CHUNKS; ++c) {
      float4 v = *(const float4*)&rowbuf[rb][c * (TPB * 4) + tid * 4];
      v.x *= 0.5f; v.y *= 0.5f; v.z *= 0.5f; v.w *= 0.5f;
      lmax = fmaxf(lmax, fmaxf(fmaxf(v.x, v.y), fmaxf(v.z, v.w)));
      xs[c] = v;
    }

    // Kick off the prefetch of the next assigned row; it streams from HBM
    // while we run ~50 VALU-bound bisection iterations on this one.
    {
      int nr = row + stride;
      if (nr < nrows)   // block-uniform condition
        asyncLoadRow(X + (long long)nr * D_DIM, rowbuf[rb ^ 1], tid);
    }

    const float mvs = blockMax(lmax, red, p, lane, wav); p ^= 1;

    // alpha = 1.5: tau_hi - tau_lo = 1 - (1/4096)^0.5 = 1 - 1/64 (exact).
    float tau_lo = mvs - 1.0f;
    float dm     = 0.984375f;

    float f_lo = blockSum(relu2Partial(xs, tau_lo, lmax), red, p, lane, wav)
                 - 1.0f;
    p ^= 1;

    float tau_m = tau_lo;
    float f_m   = f_lo;
#pragma unroll 2
    for (int it = 0; it < NITER; ++it) {
      dm *= 0.5f;
      tau_m = tau_lo + dm;
      f_m = blockSum(relu2Partial(xs, tau_m, lmax), red, p, lane, wav) - 1.0f;
      p ^= 1;
      if (f_m * f_lo >= 0.0f) tau_lo = tau_m;   // uniform: f_m identical per block
    }

    // Reference normalizes by sum(p_m) at the final tau_m, which is f_m + 1.
    const float inv = 1.0f / (f_m + 1.0f);
    float* outRow = Out + (long long)row * D_DIM;
#pragma unroll
    for (int c = 0; c < CHUNKS; ++c) {
      float a = fmaxf(xs[c].x - tau_m, 0.0f);
      float b = fmaxf(xs[c].y - tau_m, 0.0f);
      float g = fmaxf(xs[c].z - tau_m, 0.0f);
      float h = fmaxf(xs[c].w - tau_m, 0.0f);
      float4 o;
      o.x = a * a * inv; o.y = b * b * inv;
      o.z = g * g * inv; o.w = h * h * inv;
      *(float4*)(outRow + c * (TPB * 4) + tid * 4) = o;
    }
  }
}

extern "C" void kernel_launch(void* const* d_in, const int* in_sizes, int n_in,
                              void* d_out, int out_size, void* d_ws, size_t ws_size,
                              hipStream_t stream) {
  const float* X   = (const float*)d_in[0];
  float*       Out = (float*)d_out;
  const int nrows  = in_sizes[0] / D_DIM;   // 16384 for (8,2048,4096)

  int grid = nrows < 2048 ? nrows : 2048;   // grid-strided; ~8 rows per block
  if (grid < 1) grid = 1;
  entmax15_bisect_kernel<<<grid, TPB, 0, stream>>>(X, Out, nrows);
}